// CRFClassifier_61667140435973
// MI455X (gfx1250) — compile-verified
//
#include <hip/hip_runtime.h>
#include <hip/hip_bf16.h>

#define Bn 64
#define Tn 512
#define Hn 768
#define Ln 32

typedef float v2f __attribute__((ext_vector_type(2)));
typedef float v8f __attribute__((ext_vector_type(8)));

// ---------------------------------------------------------------------------
// Kernel 1: emissions = hidden_states @ W + b  via fp32 WMMA 16x16x4
// grid: 256 blocks x 256 threads (8 waves); each wave: 16 rows x 32 cols tile
// ---------------------------------------------------------------------------
__global__ void emissions_gemm(const float* __restrict__ hs,
                               const float* __restrict__ W,
                               const float* __restrict__ bias,
                               float* __restrict__ em) {
  const int wave = threadIdx.x >> 5;           // 0..7
  const int lane = threadIdx.x & 31;
  const int tile = blockIdx.x * 8 + wave;      // 2048 tiles of 16 rows
  const int row0 = tile * 16;
  const int r    = lane & 15;
  const int kk   = (lane >> 4) * 2;            // K sub-offset per A/B layout

  v8f c0 = {};  // cols 0..15
  v8f c1 = {};  // cols 16..31
  const float* arow = hs + (size_t)(row0 + r) * Hn + kk;

  for (int k0 = 0; k0 < Hn; k0 += 4) {
    v2f a = *(const v2f*)(arow + k0);          // A: row r, K = k0+kk, k0+kk+1
    v2f b0, b1;
    b0.x = W[(size_t)(k0 + kk) * Ln + r];
    b0.y = W[(size_t)(k0 + kk + 1) * Ln + r];
    b1.x = W[(size_t)(k0 + kk) * Ln + 16 + r];
    b1.y = W[(size_t)(k0 + kk + 1) * Ln + 16 + r];
    c0 = __builtin_amdgcn_wmma_f32_16x16x4_f32(false, a, false, b0,
                                               (short)0, c0, false, false);
    c1 = __builtin_amdgcn_wmma_f32_16x16x4_f32(false, a, false, b1,
                                               (short)0, c1, false, false);
  }

  const float bias0 = bias[r];
  const float bias1 = bias[16 + r];
  const int mbase = (lane >> 4) * 8;           // C/D layout: M = v + (l>>4)*8
#pragma unroll
  for (int v = 0; v < 8; ++v) {
    const int m = mbase + v;
    em[(size_t)(row0 + m) * Ln + r]      = c0[v] + bias0;
    em[(size_t)(row0 + m) * Ln + 16 + r] = c1[v] + bias1;
  }
}

// ---------------------------------------------------------------------------
// Kernel 2: CRF scans. blocks 0..63: forward/logsumexp + numerator (loss path)
//           blocks 64..127: Viterbi + LDS backtrack (prediction path)
// One wave (32 lanes) per block; lane j owns label j.
// ---------------------------------------------------------------------------
__global__ void crf_scan(const float* __restrict__ em,
                         const int* __restrict__ amask,
                         const int* __restrict__ labels,
                         const float* __restrict__ startT,
                         const float* __restrict__ endT,
                         const float* __restrict__ trans,
                         float* __restrict__ partial,   // [64] numer - denom
                         float* __restrict__ pred) {    // d_out + 1, B*T floats
  __shared__ __align__(16) float s_lds[Ln];
  __shared__ __align__(16) float p_lds[Ln];
  __shared__ unsigned char hist[(Tn - 1) * Ln];        // viterbi backpointers

  const int  b  = blockIdx.x & 63;
  const bool vit = (blockIdx.x >= 64);
  const int  j  = threadIdx.x;                         // 0..31

  // sequence length = sum(mask) (mask is monotone: t < len)
  int len = 0;
  for (int t = j; t < Tn; t += 32) len += amask[b * Tn + t];
#pragma unroll
  for (int s = 16; s; s >>= 1) len += __shfl_xor(len, s, 32);

  const float* emb = em + (size_t)b * Tn * Ln;
  float score = startT[j] + emb[j];                    // score0

  if (!vit) {
    // ---- forward algorithm with exp-trick: E[i] = exp(trans[i][j]) ----
    float E[Ln];
#pragma unroll
    for (int i = 0; i < Ln; ++i) E[i] = __expf(trans[i * Ln + j]);

    for (int t = 1; t < len; ++t) {
      s_lds[j] = score;
      __syncthreads();
      // running max over all 32 scores (b128 broadcast reads)
      const float4* s4 = (const float4*)s_lds;
      float mx = -1e30f;
#pragma unroll
      for (int i = 0; i < 8; ++i) {
        float4 v = s4[i];
        mx = fmaxf(mx, fmaxf(fmaxf(v.x, v.y), fmaxf(v.z, v.w)));
      }
      p_lds[j] = __expf(score - mx);
      __syncthreads();
      const float4* p4 = (const float4*)p_lds;
      float a0 = 0.f, a1 = 0.f, a2 = 0.f, a3 = 0.f;
#pragma unroll
      for (int i = 0; i < 8; ++i) {
        float4 v = p4[i];
        a0 = fmaf(v.x, E[4 * i + 0], a0);
        a1 = fmaf(v.y, E[4 * i + 1], a1);
        a2 = fmaf(v.z, E[4 * i + 2], a2);
        a3 = fmaf(v.w, E[4 * i + 3], a3);
      }
      score = emb[t * Ln + j] + mx + __logf((a0 + a1) + (a2 + a3));
      __syncthreads();
    }

    // denom = logsumexp_j(score + end)
    float v = score + endT[j];
    float mx = v;
#pragma unroll
    for (int s = 16; s; s >>= 1) mx = fmaxf(mx, __shfl_xor(mx, s, 32));
    float e = __expf(v - mx);
#pragma unroll
    for (int s = 16; s; s >>= 1) e += __shfl_xor(e, s, 32);
    const float denom = mx + __logf(e);

    // numerator: lanes strided over t
    const int* lab = labels + b * Tn;
    float acc = 0.f;
    for (int t = j; t < Tn; t += 32) {
      if (t >= 1 && t < len) {
        const int lt = lab[t], lp = lab[t - 1];
        acc += emb[t * Ln + lt] + trans[lp * Ln + lt];
      }
    }
#pragma unroll
    for (int s = 16; s; s >>= 1) acc += __shfl_xor(acc, s, 32);
    if (j == 0) {
      const int l0 = lab[0], ll = lab[len - 1];
      const float numer = startT[l0] + emb[l0] + acc + endT[ll];
      partial[b] = numer - denom;
    }
  } else {
    // ---- Viterbi: transitions column j in registers ----
    float tr[Ln];
#pragma unroll
    for (int i = 0; i < Ln; ++i) tr[i] = trans[i * Ln + j];

    for (int t = 1; t < len; ++t) {
      s_lds[j] = score;
      __syncthreads();
      const float4* s4 = (const float4*)s_lds;
      float best = -1e30f;
      int   barg = 0;
#pragma unroll
      for (int i = 0; i < 8; ++i) {
        float4 v = s4[i];
        float c;
        c = v.x + tr[4 * i + 0]; if (c > best) { best = c; barg = 4 * i + 0; }
        c = v.y + tr[4 * i + 1]; if (c > best) { best = c; barg = 4 * i + 1; }
        c = v.z + tr[4 * i + 2]; if (c > best) { best = c; barg = 4 * i + 2; }
        c = v.w + tr[4 * i + 3]; if (c > best) { best = c; barg = 4 * i + 3; }
      }
      hist[(t - 1) * Ln + j] = (unsigned char)barg;
      score = best + emb[t * Ln + j];
      __syncthreads();
    }

    // argmax across lanes of score + end (lowest index on ties)
    float bv = score + endT[j];
    int   bi = j;
#pragma unroll
    for (int s = 16; s; s >>= 1) {
      float ov = __shfl_xor(bv, s, 32);
      int   oi = __shfl_xor(bi, s, 32);
      if (ov > bv || (ov == bv && oi < bi)) { bv = ov; bi = oi; }
    }
    const int best_last = bi;
    const int last_idx  = len - 1;

    // PAD tail + last tag (disjoint addresses from backtrack writes)
    for (int t = j; t < Tn; t += 32) {
      if (t > last_idx)       pred[b * Tn + t] = 0.0f;           // PAD
      else if (t == last_idx) pred[b * Tn + t] = (float)best_last;
    }
    // sequential backtrack out of LDS (one lane)
    if (j == 0) {
      int carry = best_last;
      for (int t = last_idx - 1; t >= 0; --t) {
        carry = (int)hist[t * Ln + carry];   // tags[t] = hist[t][tags[t+1]]
        pred[b * Tn + t] = (float)carry;
      }
    }
  }
}

// ---------------------------------------------------------------------------
// Kernel 3: loss = -sum(partial)
// ---------------------------------------------------------------------------
__global__ void loss_reduce(const float* __restrict__ partial,
                            float* __restrict__ out) {
  __shared__ float sh[Bn];
  const int i = threadIdx.x;
  sh[i] = partial[i];
  __syncthreads();
  if (i < 32) {
    float v = sh[i] + sh[i + 32];
#pragma unroll
    for (int s = 16; s; s >>= 1) v += __shfl_xor(v, s, 32);
    if (i == 0) out[0] = -v;
  }
}

// ---------------------------------------------------------------------------
extern "C" void kernel_launch(void* const* d_in, const int* in_sizes, int n_in,
                              void* d_out, int out_size, void* d_ws, size_t ws_size,
                              hipStream_t stream) {
  const float* hs     = (const float*)d_in[0];  // (B,T,H) f32
  const int*   amask  = (const int*)d_in[1];    // (B,T) i32
  const int*   labels = (const int*)d_in[2];    // (B,T) i32
  const float* W      = (const float*)d_in[3];  // (H,L) f32
  const float* bias   = (const float*)d_in[4];  // (L,) f32
  const float* startT = (const float*)d_in[5];  // (L,) f32
  const float* endT   = (const float*)d_in[6];  // (L,) f32
  const float* trans  = (const float*)d_in[7];  // (L,L) f32

  float* out = (float*)d_out;                   // [0]=loss, [1..]=pred (B,T)
  float* em      = (float*)d_ws;                // B*T*L floats = 4 MB
  float* partial = em + (size_t)Bn * Tn * Ln;   // 64 floats

  emissions_gemm<<<256, 256, 0, stream>>>(hs, W, bias, em);
  crf_scan<<<2 * Bn, 32, 0, stream>>>(em, amask, labels, startT, endT, trans,
                                      partial, out + 1);
  loss_reduce<<<1, Bn, 0, stream>>>(partial, out);
}